// MatrixModel_18227841204531
// MI455X (gfx1250) — compile-verified
//
#include <hip/hip_runtime.h>
#include <hip/hip_bf16.h>

// out[n,i] = a[i] + t[n]*b[i], a = A@theta, b = B@theta
// N = 16384, I = 512, K = 10. Output (N, I) row-major f32.

typedef __attribute__((ext_vector_type(2))) float v2f;
typedef __attribute__((ext_vector_type(8))) float v8f;

#define N_TOTAL 16384
#define I_TOTAL 512
#define K_DIM   10
#define NT_PER_WAVE 8   // n-tiles (16 rows each) handled per wave

__global__ __launch_bounds__(256) void MatrixModel_wmma_kernel(
    const float* __restrict__ t,      // (16384)
    const float* __restrict__ A,      // (512, 10)
    const float* __restrict__ B,      // (512, 10)
    const float* __restrict__ theta,  // (10, 1)
    float* __restrict__ out)          // (16384, 512)
{
    const int lane = threadIdx.x & 31;
    const int wave = threadIdx.x >> 5;
    const int gw   = blockIdx.x * 8 + wave;       // global wave id, 0..4095

    const int i_tile = gw & 31;                   // 32 column tiles of 16
    const int n_grp  = gw >> 5;                   // 0..127 row groups (8 tiles each)
    const int i0     = i_tile * 16;
    const int colLane = lane & 15;
    const bool lo = (lane < 16);

    // ---- inline a[i] = A[i,:]@theta, b[i] = B[i,:]@theta for this wave's 16 columns ----
    const float* Arow = A + (size_t)(i0 + colLane) * K_DIM;
    const float* Brow = B + (size_t)(i0 + colLane) * K_DIM;
    float av = 0.0f, bv = 0.0f;
#pragma unroll
    for (int k = 0; k < K_DIM; ++k) {
        const float th = theta[k];          // uniform -> scalar loads
        av = fmaf(Arow[k], th, av);
        bv = fmaf(Brow[k], th, bv);
    }

    // ---- B operand (4x16, KxN): row K=0 = b[i], row K=1 = a[i], rows K=2,3 = 0 ----
    // Layout: lanes 0-15 hold N=lane with VGPR0->K=0, VGPR1->K=1;
    //         lanes 16-31 hold N=lane-16 with VGPR0->K=2, VGPR1->K=3 (zeros).
    v2f Bop;
    Bop.x = lo ? bv : 0.0f;
    Bop.y = lo ? av : 0.0f;

    // ---- loop over 8 n-tiles; one V_WMMA_F32_16X16X4_F32 per 16x16 output tile ----
    for (int tt = 0; tt < NT_PER_WAVE; ++tt) {
        const int n0 = (n_grp * NT_PER_WAVE + tt) * 16;

        // A operand (16x4, MxK): row M = n0+m has [t, 1, 0, 0].
        // Layout: lanes 0-15 hold M=lane with VGPR0->K=0, VGPR1->K=1;
        //         lanes 16-31 hold M=lane-16 with VGPR0->K=2, VGPR1->K=3 (zeros).
        const float tv = t[n0 + colLane];   // in-range load for all lanes
        v2f Aop;
        Aop.x = lo ? tv   : 0.0f;
        Aop.y = lo ? 1.0f : 0.0f;

        v8f c = {};
        // D = A x B + C : D[m,j] = t[n0+m]*b[i0+j] + a[i0+j]
        c = __builtin_amdgcn_wmma_f32_16x16x4_f32(
                /*neg_a=*/false, Aop,
                /*neg_b=*/false, Bop,
                /*c_mod=*/(short)0, c,
                /*reuse_a=*/false, /*reuse_b=*/false);

        // D layout (16x16 f32): VGPR r -> row r (lanes 0-15) / row r+8 (lanes 16-31),
        // column = lane & 15.
        const int rowOff = (lane >> 4) << 3;              // 0 or 8
        float* outp = out + (size_t)(n0 + rowOff) * I_TOTAL + i0 + colLane;
#pragma unroll
        for (int r = 0; r < 8; ++r) {
            outp[(size_t)r * I_TOTAL] = c[r];
        }
    }
}

extern "C" void kernel_launch(void* const* d_in, const int* in_sizes, int n_in,
                              void* d_out, int out_size, void* d_ws, size_t ws_size,
                              hipStream_t stream) {
    const float* t     = (const float*)d_in[0];   // (16384)
    const float* A     = (const float*)d_in[1];   // (512, 10)
    const float* B     = (const float*)d_in[2];   // (512, 10)
    const float* theta = (const float*)d_in[3];   // (10, 1)
    float* out = (float*)d_out;                   // (16384, 512)

    // 32768 tiles / 8 tiles-per-wave = 4096 waves; 8 waves per 256-thread block.
    dim3 grid(512), block(256);
    MatrixModel_wmma_kernel<<<grid, block, 0, stream>>>(t, A, B, theta, out);
}